// Net_6682969113359
// MI455X (gfx1250) — compile-verified
//
#include <hip/hip_runtime.h>
#include <hip/hip_bf16.h>
#include <cmath>

// ---------------------------------------------------------------------------
// Dense dilated-conv chain (DenseNet-style), B=16, H=W=256, fp32.
//   - conv chain: memory-bound (whole working set ~76MB fits MI455X 192MB L2),
//     done with per-pixel VALU FMA (WMMA Toeplitz expansion would be <20%
//     MAC-utilization and add traffic to a bandwidth-bound problem).
//   - FC head: exact-fp32 16x10x18 GEMM via V_WMMA_F32_16X16X4_F32 (5 k-steps).
// ---------------------------------------------------------------------------

#define BATCH 16
#define HDIM  256
#define WDIM  256
#define IMG   (HDIM * WDIM)   // 65536 pixels per plane
#define NLAYERS 18

typedef __attribute__((ext_vector_type(2))) float v2f;
typedef __attribute__((ext_vector_type(8))) float v8f;

// ---- Layer kernel: out = relu(conv_dilated(concat(x, layers[0..nch-2]), w) + bias)
// grid = (HDIM rows, BATCH images), block = WDIM threads (one output row).
// Also produces a deterministic per-row partial sum for the spatial mean.
__global__ __launch_bounds__(WDIM)
void conv_layer(const float* __restrict__ x,       // [B][H][W]
                const float* __restrict__ layers,  // ws: [18][B][H][W]
                const float* __restrict__ w,       // [nch][3][3]
                const float* __restrict__ bias,    // [1]
                float* __restrict__ out,           // this layer's plane [B][H][W]
                float* __restrict__ partial,       // [B][H] row sums for this layer
                int nch, int dil)
{
    const int xc = threadIdx.x;        // column
    const int y  = blockIdx.x;         // row
    const int b  = blockIdx.y;         // image

    const int ym = y - dil, yp = y + dil;
    const int xm = xc - dil, xp = xc + dil;
    const bool ymOK = (ym >= 0), ypOK = (yp < HDIM);
    const bool xmOK = (xm >= 0), xpOK = (xp < WDIM);

    float acc = bias[0];
    for (int c = 0; c < nch; ++c) {
        const float* src = (c == 0) ? (x + (size_t)b * IMG)
                                    : (layers + ((size_t)(c - 1) * BATCH + b) * IMG);
        const float* wc = w + c * 9;   // uniform -> scalar loads
        if (ymOK) {
            const float* r = src + ym * WDIM;
            if (xmOK) acc += wc[0] * r[xm];
            acc += wc[1] * r[xc];
            if (xpOK) acc += wc[2] * r[xp];
        }
        {
            const float* r = src + y * WDIM;
            if (xmOK) acc += wc[3] * r[xm];
            acc += wc[4] * r[xc];
            if (xpOK) acc += wc[5] * r[xp];
        }
        if (ypOK) {
            const float* r = src + yp * WDIM;
            if (xmOK) acc += wc[6] * r[xm];
            acc += wc[7] * r[xc];
            if (xpOK) acc += wc[8] * r[xp];
        }
    }
    float v = fmaxf(acc, 0.0f);
    out[(size_t)b * IMG + y * WDIM + xc] = v;

    // Deterministic row-sum reduction (no float atomics -> replay-stable).
    __shared__ float red[WDIM];
    red[xc] = v;
    __syncthreads();
    for (int s = WDIM / 2; s > 0; s >>= 1) {
        if (xc < s) red[xc] += red[xc + s];
        __syncthreads();
    }
    if (xc == 0) partial[b * HDIM + y] = red[0];
}

// ---- Fold per-row partials into feats[b][layer] = mean over HxW.
// grid = (18 layers, 16 images), block = HDIM threads.
__global__ __launch_bounds__(HDIM)
void reduce_partials(const float* __restrict__ partial,  // [18][B][H]
                     float* __restrict__ feats)          // [B][18]
{
    const int layer = blockIdx.x, b = blockIdx.y, t = threadIdx.x;
    __shared__ float red[HDIM];
    red[t] = partial[((size_t)layer * BATCH + b) * HDIM + t];
    __syncthreads();
    for (int s = HDIM / 2; s > 0; s >>= 1) {
        if (t < s) red[t] += red[t + s];
        __syncthreads();
    }
    if (t == 0) feats[b * NLAYERS + layer] = red[0] * (1.0f / (float)IMG);
}

// ---- FC + log_softmax on one wave via V_WMMA_F32_16X16X4_F32.
// D[16,16] = feats[16,18] x fc_w^T (N padded 10->16, K padded 18->20).
// A frag (16x4 f32): lane m (0..15) holds K=k0,k0+1; lane m+16 holds K=k0+2,k0+3.
// B frag (4x16 f32): lane n (0..15) holds K rows k0,k0+1; lane n+16 holds k0+2,k0+3.
// D: VGPR r, lanes 0..15 -> row r; lanes 16..31 -> row r+8.
__global__ __launch_bounds__(32)
void fc_logsoftmax(const float* __restrict__ feats,  // [16][18]
                   const float* __restrict__ fcw,    // [10][18]
                   const float* __restrict__ fcb,    // [10]
                   float* __restrict__ outp)         // [16][10]
{
    const int lane = threadIdx.x;
    const int hf   = lane >> 4;       // 0 or 1 (K sub-pair select)
    const int l15  = lane & 15;       // M (for A) / N (for B) index

    v8f acc = {};
#pragma unroll
    for (int t = 0; t < 5; ++t) {
        const int k0 = 4 * t + 2 * hf;
        v2f a, bm;
        a.x = (k0     < NLAYERS) ? feats[l15 * NLAYERS + k0]     : 0.0f;
        a.y = (k0 + 1 < NLAYERS) ? feats[l15 * NLAYERS + k0 + 1] : 0.0f;
        const bool nOK = (l15 < 10);
        bm.x = (nOK && k0     < NLAYERS) ? fcw[l15 * NLAYERS + k0]     : 0.0f;
        bm.y = (nOK && k0 + 1 < NLAYERS) ? fcw[l15 * NLAYERS + k0 + 1] : 0.0f;
        // (neg_a, A, neg_b, B, c_mod, C, reuse_a, reuse_b)
        acc = __builtin_amdgcn_wmma_f32_16x16x4_f32(
            false, a, false, bm, (short)0, acc, false, false);
    }

    __shared__ float lg[16 * 16];
#pragma unroll
    for (int r = 0; r < 8; ++r) {
        const int row = r + 8 * hf;
        lg[row * 16 + l15] = acc[r];
    }
    __syncthreads();

    if (lane < 16) {
        float v[10];
        float m = -INFINITY;
#pragma unroll
        for (int j = 0; j < 10; ++j) {
            v[j] = lg[lane * 16 + j] + fcb[j];
            m = fmaxf(m, v[j]);
        }
        float s = 0.0f;
#pragma unroll
        for (int j = 0; j < 10; ++j) s += expf(v[j] - m);
        const float ls = logf(s);
#pragma unroll
        for (int j = 0; j < 10; ++j) outp[lane * 10 + j] = v[j] - m - ls;
    }
}

extern "C" void kernel_launch(void* const* d_in, const int* in_sizes, int n_in,
                              void* d_out, int out_size, void* d_ws, size_t ws_size,
                              hipStream_t stream) {
    (void)in_sizes; (void)n_in; (void)out_size; (void)ws_size;
    static const int DIL[NLAYERS] = {1,1,1,2,2,2,3,3,3,1,1,1,2,2,2,3,3,3};

    // Input order: x, conv_ws[0..17], conv_bs[0..17], fc_w, fc_b
    const float* x = (const float*)d_in[0];

    // Workspace layout: [18 layer planes][row partials][feats]
    float* layers   = (float*)d_ws;                                   // 18*16*65536 f32 (72MB)
    float* partials = layers + (size_t)NLAYERS * BATCH * IMG;          // 18*16*256 f32
    float* feats    = partials + (size_t)NLAYERS * BATCH * HDIM;       // 16*18 f32

    for (int i = 0; i < NLAYERS; ++i) {
        conv_layer<<<dim3(HDIM, BATCH), WDIM, 0, stream>>>(
            x, layers,
            (const float*)d_in[1 + i],            // conv_ws[i]: (i+1)*9
            (const float*)d_in[19 + i],           // conv_bs[i]: 1
            layers + (size_t)i * BATCH * IMG,     // this layer's plane
            partials + (size_t)i * BATCH * HDIM,  // this layer's row sums
            i + 1, DIL[i]);
    }

    reduce_partials<<<dim3(NLAYERS, BATCH), HDIM, 0, stream>>>(partials, feats);

    fc_logsoftmax<<<1, 32, 0, stream>>>(
        feats, (const float*)d_in[37], (const float*)d_in[38], (float*)d_out);
}